// ENC_DEC_89429809037334
// MI455X (gfx1250) — compile-verified
//
#include <hip/hip_runtime.h>
#include <math.h>

typedef __attribute__((ext_vector_type(2))) float v2f;
typedef __attribute__((ext_vector_type(8))) float v8f;

__device__ __forceinline__ void atomAddF(float* p, float v) {
  // Hardware f32 atomic add (global_atomic_add_f32), no CAS loop.
  unsafeAtomicAdd(p, v);
}

// ---------------------------------------------------------------------------
// Degree / norm precompute
// ---------------------------------------------------------------------------
__global__ void deg_scatter_kernel(const int* __restrict__ col,
                                   const float* __restrict__ ew,
                                   float* __restrict__ deg, int E) {
  int e = blockIdx.x * blockDim.x + threadIdx.x;
  if (e < E) atomAddF(&deg[col[e]], ew[e]);
}

__global__ void deg_finalize_kernel(float* __restrict__ deg, int N) {
  int i = blockIdx.x * blockDim.x + threadIdx.x;
  if (i < N) {
    float d = deg[i] + 1.0f;           // self-loop weight 1
    deg[i] = (d > 0.0f) ? rsqrtf(d) : 0.0f;   // now holds D^-1/2
  }
}

__global__ void norm_kernel(const int* __restrict__ row, const int* __restrict__ col,
                            const float* __restrict__ ew, const float* __restrict__ dis,
                            float* __restrict__ nrm, int E) {
  int e = blockIdx.x * blockDim.x + threadIdx.x;
  if (e < E) nrm[e] = dis[row[e]] * ew[e] * dis[col[e]];
}

// ---------------------------------------------------------------------------
// GEMM: Y[N x FO] = X[N x 128] @ W[128 x FO], f32 WMMA 16x16x4.
// One wave owns a 16-row block and ALL FO output columns (NT tiles of 16).
// N must be a multiple of 16 (100000 = 16 * 6250).
// ---------------------------------------------------------------------------
template <int NT>   // NT = FO/16 : 8 for FO=128, 4 for FO=64
__global__ void gemm_wmma_kernel(const float* __restrict__ X,
                                 const float* __restrict__ W,
                                 float* __restrict__ Y, int nRowBlocks) {
  constexpr int K = 128;
  constexpr int FO = NT * 16;
  const int wavesPerBlock = blockDim.x >> 5;
  const int wave = blockIdx.x * wavesPerBlock + (threadIdx.x >> 5);
  if (wave >= nRowBlocks) return;
  const int lane = threadIdx.x & 31;
  const int hi = lane >> 4;    // 0: K-pair {k,k+1}, 1: {k+2,k+3}
  const int lo = lane & 15;    // M (for A) / N (for B) within tile
  const int rowBase = wave * 16;
  const float* __restrict__ Xrow = X + (size_t)(rowBase + lo) * K;

  v8f acc[NT];
#pragma unroll
  for (int t = 0; t < NT; ++t) acc[t] = (v8f){0.f,0.f,0.f,0.f,0.f,0.f,0.f,0.f};

  for (int k = 0; k < K; k += 4) {
    const int ka = k + 2 * hi;
    v2f a;
    a.x = Xrow[ka];
    a.y = Xrow[ka + 1];
#pragma unroll
    for (int t = 0; t < NT; ++t) {
      const int n = t * 16 + lo;
      v2f b;
      b.x = W[(size_t)ka * FO + n];
      b.y = W[(size_t)(ka + 1) * FO + n];
      acc[t] = __builtin_amdgcn_wmma_f32_16x16x4_f32(
          false, a, false, b, (short)0, acc[t], false, false);
    }
  }

#pragma unroll
  for (int t = 0; t < NT; ++t) {
    const int coln = t * 16 + lo;
#pragma unroll
    for (int v = 0; v < 8; ++v) {
      const int r = rowBase + v + 8 * hi;   // C/D layout: VGPR v -> M=v / v+8
      Y[(size_t)r * FO + coln] = acc[t][v];
    }
  }
}

// ---------------------------------------------------------------------------
// Fused edge scatter: one wave per edge (grid-stride). Reads row/col/norm once,
// scatters NMAT feature vectors with f32 atomics (L2-resident destinations).
// ---------------------------------------------------------------------------
template <int NMAT, int F>
__global__ void scatter_kernel(const int* __restrict__ row, const int* __restrict__ col,
                               const float* __restrict__ nrm,
                               const float* __restrict__ t0, const float* __restrict__ t1,
                               const float* __restrict__ t2,
                               float* __restrict__ a0, float* __restrict__ a1,
                               float* __restrict__ a2, int E) {
  constexpr int VPL = F / 32;   // floats per lane: 4 (F=128) or 2 (F=64)
  const int lane = threadIdx.x & 31;
  int wave = (blockIdx.x * blockDim.x + threadIdx.x) >> 5;
  const int nw = (gridDim.x * blockDim.x) >> 5;
  for (int e = wave; e < E; e += nw) {
    const int r = row[e];
    const int c = col[e];
    const float w = nrm[e];
    const size_t rb = (size_t)r * F + lane * VPL;
    const size_t cb = (size_t)c * F + lane * VPL;
    {
      float v[VPL];
#pragma unroll
      for (int j = 0; j < VPL; ++j) v[j] = t0[rb + j] * w;
#pragma unroll
      for (int j = 0; j < VPL; ++j) atomAddF(a0 + cb + j, v[j]);
    }
    if (NMAT > 1) {
      float v[VPL];
#pragma unroll
      for (int j = 0; j < VPL; ++j) v[j] = t1[rb + j] * w;
#pragma unroll
      for (int j = 0; j < VPL; ++j) atomAddF(a1 + cb + j, v[j]);
    }
    if (NMAT > 2) {
      float v[VPL];
#pragma unroll
      for (int j = 0; j < VPL; ++j) v[j] = t2[rb + j] * w;
#pragma unroll
      for (int j = 0; j < VPL; ++j) atomAddF(a2 + cb + j, v[j]);
    }
  }
}

// ---------------------------------------------------------------------------
// Finalize: add self-loop (t*dis^2) + bias, tanh, elementwise product.
// ---------------------------------------------------------------------------
__global__ void finalize3_kernel(const float* __restrict__ a0, const float* __restrict__ a1,
                                 const float* __restrict__ a2,
                                 const float* __restrict__ t0, const float* __restrict__ t1,
                                 const float* __restrict__ t2,
                                 const float* __restrict__ b0, const float* __restrict__ b1,
                                 const float* __restrict__ b2,
                                 const float* __restrict__ dis, float* __restrict__ out,
                                 int N) {
  const size_t idx = (size_t)blockIdx.x * blockDim.x + threadIdx.x;
  const size_t total = (size_t)N * 128;
  if (idx >= total) return;
  const int i = (int)(idx >> 7);
  const int f = (int)(idx & 127);
  const float d = dis[i];
  const float d2 = d * d;
  const float x0 = tanhf(a0[idx] + t0[idx] * d2 + b0[f]);
  const float x1 = tanhf(a1[idx] + t1[idx] * d2 + b1[f]);
  const float x2 = tanhf(a2[idx] + t2[idx] * d2 + b2[f]);
  out[idx] = x0 * x1 * x2;
}

__global__ void finalize2_kernel(const float* __restrict__ a0, const float* __restrict__ a1,
                                 const float* __restrict__ t0, const float* __restrict__ t1,
                                 const float* __restrict__ b0, const float* __restrict__ b1,
                                 const float* __restrict__ dis, float* __restrict__ out,
                                 int N) {
  const size_t idx = (size_t)blockIdx.x * blockDim.x + threadIdx.x;
  const size_t total = (size_t)N * 128;
  if (idx >= total) return;
  const int i = (int)(idx >> 7);
  const int f = (int)(idx & 127);
  const float d = dis[i];
  const float d2 = d * d;
  const float x0 = tanhf(a0[idx] + t0[idx] * d2 + b0[f]);
  const float x1 = tanhf(a1[idx] + t1[idx] * d2 + b1[f]);
  out[idx] = x0 * x1;
}

__global__ void finalize1_kernel(const float* __restrict__ a0, const float* __restrict__ t0,
                                 const float* __restrict__ b0,
                                 const float* __restrict__ dis, float* __restrict__ out,
                                 int N) {
  const size_t idx = (size_t)blockIdx.x * blockDim.x + threadIdx.x;
  const size_t total = (size_t)N * 64;
  if (idx >= total) return;
  const int i = (int)(idx >> 6);
  const int f = (int)(idx & 63);
  const float d = dis[i];
  out[idx] = tanhf(a0[idx] + t0[idx] * d * d + b0[f]);
}

// ---------------------------------------------------------------------------
// Host launcher
// ---------------------------------------------------------------------------
extern "C" void kernel_launch(void* const* d_in, const int* in_sizes, int n_in,
                              void* d_out, int out_size, void* d_ws, size_t ws_size,
                              hipStream_t stream) {
  const int K = 128;
  const int E = in_sizes[2];              // edge_weight count
  const int N = in_sizes[0] / K;          // h0 is [N,128]

  const float* h0 = (const float*)d_in[0];
  const int*   ei = (const int*)d_in[1];
  const float* ew = (const float*)d_in[2];
  const float* W1 = (const float*)d_in[3];  const float* b1 = (const float*)d_in[4];
  const float* W2 = (const float*)d_in[5];  const float* b2 = (const float*)d_in[6];
  const float* W3 = (const float*)d_in[7];  const float* b3 = (const float*)d_in[8];
  const float* W4 = (const float*)d_in[9];  const float* b4 = (const float*)d_in[10];
  const float* W5 = (const float*)d_in[11]; const float* b5 = (const float*)d_in[12];
  const float* W6 = (const float*)d_in[13]; const float* b6 = (const float*)d_in[14];
  const int* row = ei;
  const int* col = ei + E;
  float* out = (float*)d_out;

  // Workspace carve-up (256B aligned)
  char* ws = (char*)d_ws;
  size_t o = 0;
  auto alloc = [&](size_t bytes) -> float* {
    float* p = (float*)(ws + o);
    o += (bytes + 255) & ~(size_t)255;
    return p;
  };
  const size_t bigBytes = (size_t)N * K * sizeof(float);   // 51.2 MB
  float* dis  = alloc((size_t)N * sizeof(float));
  float* nrm  = alloc((size_t)E * sizeof(float));
  float* tmp1 = alloc(bigBytes);   // also h123 later
  float* tmp2 = alloc(bigBytes);   // also h45 later
  float* tmp3 = alloc(bigBytes);   // also tmp6 (N x 64) later
  float* agg1 = alloc(bigBytes);
  float* agg2 = alloc(bigBytes);
  float* agg3 = alloc(bigBytes);

  const int T = 256;
  const int eBlocks = (E + T - 1) / T;
  const int nBlocks = (N + T - 1) / T;
  const int rowBlocks = N / 16;                       // 6250
  const int gemmBlocks = (rowBlocks + 7) / 8;         // 8 waves/block
  const int scatBlocks = 4096;                        // grid-stride over edges
  const int f128Blocks = (int)(((size_t)N * 128 + T - 1) / T);
  const int f64Blocks  = (int)(((size_t)N * 64 + T - 1) / T);

  // --- degrees + norms ---
  hipMemsetAsync(dis, 0, (size_t)N * sizeof(float), stream);
  deg_scatter_kernel<<<eBlocks, T, 0, stream>>>(col, ew, dis, E);
  deg_finalize_kernel<<<nBlocks, T, 0, stream>>>(dis, N);
  norm_kernel<<<eBlocks, T, 0, stream>>>(row, col, ew, dis, nrm, E);

  // --- stage 1: h1,h2,h3 from h0 ---
  gemm_wmma_kernel<8><<<gemmBlocks, T, 0, stream>>>(h0, W1, tmp1, rowBlocks);
  gemm_wmma_kernel<8><<<gemmBlocks, T, 0, stream>>>(h0, W2, tmp2, rowBlocks);
  gemm_wmma_kernel<8><<<gemmBlocks, T, 0, stream>>>(h0, W3, tmp3, rowBlocks);
  hipMemsetAsync(agg1, 0, bigBytes, stream);
  hipMemsetAsync(agg2, 0, bigBytes, stream);
  hipMemsetAsync(agg3, 0, bigBytes, stream);
  scatter_kernel<3, 128><<<scatBlocks, T, 0, stream>>>(row, col, nrm,
      tmp1, tmp2, tmp3, agg1, agg2, agg3, E);
  // h123 -> tmp1 (elementwise, same-index read/write is safe)
  finalize3_kernel<<<f128Blocks, T, 0, stream>>>(agg1, agg2, agg3,
      tmp1, tmp2, tmp3, b1, b2, b3, dis, tmp1, N);

  // --- stage 2: h4,h5 from h123 (tmp1) ---
  gemm_wmma_kernel<8><<<gemmBlocks, T, 0, stream>>>(tmp1, W4, tmp2, rowBlocks);
  gemm_wmma_kernel<8><<<gemmBlocks, T, 0, stream>>>(tmp1, W5, tmp3, rowBlocks);
  hipMemsetAsync(agg1, 0, bigBytes, stream);
  hipMemsetAsync(agg2, 0, bigBytes, stream);
  scatter_kernel<2, 128><<<scatBlocks, T, 0, stream>>>(row, col, nrm,
      tmp2, tmp3, nullptr, agg1, agg2, nullptr, E);
  // h45 -> tmp2
  finalize2_kernel<<<f128Blocks, T, 0, stream>>>(agg1, agg2,
      tmp2, tmp3, b4, b5, dis, tmp2, N);

  // --- stage 3: h6 from h45 (tmp2), F_out = 64 ---
  gemm_wmma_kernel<4><<<gemmBlocks, T, 0, stream>>>(tmp2, W6, tmp3, rowBlocks);
  hipMemsetAsync(agg1, 0, (size_t)N * 64 * sizeof(float), stream);
  scatter_kernel<1, 64><<<scatBlocks, T, 0, stream>>>(row, col, nrm,
      tmp3, nullptr, nullptr, agg1, nullptr, nullptr, E);
  finalize1_kernel<<<f64Blocks, T, 0, stream>>>(agg1, tmp3, b6, dis, out, N);
}